// AU_Net_13649406067417
// MI455X (gfx1250) — compile-verified
//
#include <hip/hip_runtime.h>
#include <hip/hip_bf16.h>

typedef __attribute__((ext_vector_type(2))) float v2f;
typedef __attribute__((ext_vector_type(8))) float v8f;

// ---------------------------------------------------------------------------
// One concat-segment of the K loop, fully unrolled at compile time.
//  A-fragment layout (16x4 f32): v0 -> K=2*half, v1 -> K=2*half+1, M=lane.
//  B-fragment layout (4x16 f32): v0 -> K=2*half, v1 -> K=2*half+1, N=lane.
// ---------------------------------------------------------------------------
template<int SEGW, int FOUT, bool HASADD>
__device__ __forceinline__ void gemm_segment(const float* __restrict__ Ap,
                                             const float* __restrict__ Aadd,
                                             const float* __restrict__ W,
                                             int kbase, const int ar[4],
                                             int half, int n, bool nok,
                                             v8f* acc)
{
#pragma unroll
    for (int k0 = 0; k0 < SEGW; k0 += 4) {
        const int kk = k0 + 2 * half;          // K within this segment
        const int kw = kbase + kk;             // K within W
        v2f b;
        b.x = nok ? W[(size_t)kw       * FOUT + n] : 0.0f;
        b.y = nok ? W[(size_t)(kw + 1) * FOUT + n] : 0.0f;
#pragma unroll
        for (int t = 0; t < 4; ++t) {
            const float* row = Ap + (size_t)ar[t] * SEGW;
            float ax = row[kk];
            float ay = row[kk + 1];
            if (HASADD) {
                const float* row2 = Aadd + (size_t)ar[t] * SEGW;
                ax += row2[kk];
                ay += row2[kk + 1];
            }
            v2f a = {ax, ay};
            acc[t] = __builtin_amdgcn_wmma_f32_16x16x4_f32(
                         false, a, false, b, (short)0, acc[t], false, false);
        }
    }
}

// ---------------------------------------------------------------------------
// out[N,FOUT] = act( cat(A0,A1,A2)(+Aadd on A0 seg) @ W + bias )
// One wave per block; each wave computes a 64x16 output tile (4 M-tiles).
// ---------------------------------------------------------------------------
template<int S0, int S1, int S2, int FOUT, bool RELU, bool HASADD>
__global__ __launch_bounds__(32)
void gemm_wmma(const float* __restrict__ A0, const float* __restrict__ A1,
               const float* __restrict__ A2, const float* __restrict__ Aadd,
               const float* __restrict__ W,  const float* __restrict__ bias,
               float* __restrict__ out, int N)
{
    const int l    = (int)threadIdx.x;          // 0..31
    const int lane = l & 15;
    const int half = l >> 4;
    const int row0 = (int)blockIdx.x * 64;
    const int n0   = (int)blockIdx.y * 16;
    const int n    = n0 + lane;
    const bool nok = (FOUT % 16 == 0) || (n < FOUT);

    int ar[4];
#pragma unroll
    for (int t = 0; t < 4; ++t) {
        int r = row0 + 16 * t + lane;
        ar[t] = (r < N) ? r : (N - 1);          // clamp; stores guarded below
    }

    v8f acc[4];
#pragma unroll
    for (int t = 0; t < 4; ++t)
#pragma unroll
        for (int r = 0; r < 8; ++r) acc[t][r] = 0.0f;

    gemm_segment<S0, FOUT, HASADD>(A0, Aadd, W, 0, ar, half, n, nok, acc);
    if constexpr (S1 > 0)
        gemm_segment<S1, FOUT, false>(A1, nullptr, W, S0, ar, half, n, nok, acc);
    if constexpr (S2 > 0)
        gemm_segment<S2, FOUT, false>(A2, nullptr, W, S0 + S1, ar, half, n, nok, acc);

    const float bv = (nok && bias != nullptr) ? bias[n] : 0.0f;
#pragma unroll
    for (int t = 0; t < 4; ++t) {
#pragma unroll
        for (int r = 0; r < 8; ++r) {
            const int m = row0 + 16 * t + r + 8 * half;   // D layout: VGPR r -> M=r+8*half
            float v = acc[t][r] + bv;
            if (RELU) v = fmaxf(v, 0.0f);
            if (nok && m < N) out[(size_t)m * FOUT + n] = v;
        }
    }
}

// ---------------------------------------------------------------------------
// GCN aggregation helpers
// ---------------------------------------------------------------------------
__global__ void deg_init_kernel(float* __restrict__ deg, int N) {
    int i = blockIdx.x * blockDim.x + threadIdx.x;
    if (i < N) deg[i] = 1.0f;                    // self-loop contribution
}

__global__ void deg_accum_kernel(const int* __restrict__ dst,
                                 float* __restrict__ deg, int E) {
    int e = blockIdx.x * blockDim.x + threadIdx.x;
    if (e < E) atomicAdd(&deg[dst[e]], 1.0f);
}

__global__ void dinv_kernel(const float* __restrict__ deg,
                            float* __restrict__ dinv, int N) {
    int i = blockIdx.x * blockDim.x + threadIdx.x;
    if (i < N) dinv[i] = rsqrtf(deg[i]);         // deg >= 1 always
}

// agg[n,:] = h[n,:] * dinv[n]^2   (self-loop term; also zero-inits accumulator)
__global__ void selfloop_init_kernel(const float* __restrict__ h,
                                     const float* __restrict__ dinv,
                                     float* __restrict__ agg, int total) {
    int i = blockIdx.x * blockDim.x + threadIdx.x;
    if (i < total) {
        int node = i >> 7;                       // F = 128
        float w = dinv[node];
        agg[i] = h[i] * (w * w);
    }
}

// 32 lanes per edge; each lane moves 4 floats: agg[dst] += h[src]*dinv[s]*dinv[d]
__global__ void edge_scatter_kernel(const int* __restrict__ src,
                                    const int* __restrict__ dst,
                                    const float* __restrict__ dinv,
                                    const float* __restrict__ h,
                                    float* __restrict__ agg, int E) {
    int gid  = blockIdx.x * blockDim.x + threadIdx.x;
    int e    = gid >> 5;
    int lane = gid & 31;
    if (e >= E) return;
    const int s = src[e];
    const int d = dst[e];
    const float w = dinv[s] * dinv[d];
    const float4 hv = *(const float4*)(h + (size_t)s * 128 + lane * 4);
    float* ap = agg + (size_t)d * 128 + lane * 4;
    atomicAdd(ap + 0, hv.x * w);
    atomicAdd(ap + 1, hv.y * w);
    atomicAdd(ap + 2, hv.z * w);
    atomicAdd(ap + 3, hv.w * w);
}

// x[i] = relu(x[i] + bias[i % F])   (in place)
__global__ void bias_relu_kernel(const float* __restrict__ bias,
                                 float* __restrict__ x, int total, int F) {
    int i = blockIdx.x * blockDim.x + threadIdx.x;
    if (i < total) x[i] = fmaxf(x[i] + bias[i % F], 0.0f);
}

// ---------------------------------------------------------------------------
// Host-side pipeline
// ---------------------------------------------------------------------------
extern "C" void kernel_launch(void* const* d_in, const int* in_sizes, int n_in,
                              void* d_out, int out_size, void* d_ws, size_t ws_size,
                              hipStream_t stream) {
    const float* x   = (const float*)d_in[0];
    const float* gx  = (const float*)d_in[1];
    const int*   ei  = (const int*)  d_in[2];
    const float* W1  = (const float*)d_in[3];  const float* b1  = (const float*)d_in[4];
    const float* Wdr = (const float*)d_in[5];  const float* bdr = (const float*)d_in[6];
    const float* Wg1 = (const float*)d_in[7];  const float* bg1 = (const float*)d_in[8];
    const float* Wg2 = (const float*)d_in[9];  const float* bg2 = (const float*)d_in[10];
    const float* W2  = (const float*)d_in[11]; const float* b2  = (const float*)d_in[12];
    const float* W3  = (const float*)d_in[13]; const float* b3  = (const float*)d_in[14];
    const float* Wo  = (const float*)d_in[15]; const float* bo  = (const float*)d_in[16];

    const int N = in_sizes[0] / 128;
    const int E = in_sizes[2] / 2;
    const int* src = ei;          // edge_index[0]
    const int* dst = ei + E;      // edge_index[1]

    // workspace layout (all feature buffers [N,128] f32)
    char* ws = (char*)d_ws;
    const size_t feat = (size_t)N * 128 * sizeof(float);
    float* z    = (float*)(ws + 0 * feat);          // relu(cat(x,gx)@W1+b1)
    float* z0   = (float*)(ws + 1 * feat);          // z@Wdr+bdr
    float* h    = (float*)(ws + 2 * feat);          // pre-agg h; reused as z3
    float* z1   = (float*)(ws + 3 * feat);          // agg1 -> z1; reused as z4 [N,64]
    float* z2   = (float*)(ws + 4 * feat);          // agg2 -> z2
    float* deg  = (float*)(ws + 5 * feat);
    float* dinv = (float*)(ws + 5 * feat + (size_t)N * sizeof(float));

    const int rt = (N + 63) / 64;                   // 64-row tiles per wave
    const dim3 wblk(32);
    const int NF = N * 128;
    const int eb = ((E * 32) + 255) / 256;

    // z = relu(cat(x,gx) @ W1 + b1)
    gemm_wmma<128,128,0,128,true,false><<<dim3(rt, 8), wblk, 0, stream>>>(
        x, gx, nullptr, nullptr, W1, b1, z, N);
    // z0 = z @ Wdr + bdr
    gemm_wmma<128,0,0,128,false,false><<<dim3(rt, 8), wblk, 0, stream>>>(
        z, nullptr, nullptr, nullptr, Wdr, bdr, z0, N);

    // degree / normalization (shared by both convs)
    deg_init_kernel<<<(N + 255) / 256, 256, 0, stream>>>(deg, N);
    deg_accum_kernel<<<(E + 255) / 256, 256, 0, stream>>>(dst, deg, E);
    dinv_kernel<<<(N + 255) / 256, 256, 0, stream>>>(deg, dinv, N);

    // ---- GCN conv 1: z1 = relu(Agg((z+gx)@Wg1) + bg1) ----
    gemm_wmma<128,0,0,128,false,true><<<dim3(rt, 8), wblk, 0, stream>>>(
        z, nullptr, nullptr, gx, Wg1, nullptr, h, N);
    selfloop_init_kernel<<<(NF + 255) / 256, 256, 0, stream>>>(h, dinv, z1, NF);
    edge_scatter_kernel<<<eb, 256, 0, stream>>>(src, dst, dinv, h, z1, E);
    bias_relu_kernel<<<(NF + 255) / 256, 256, 0, stream>>>(bg1, z1, NF, 128);

    // ---- GCN conv 2: z2 = relu(Agg(z1@Wg2) + bg2) ----
    gemm_wmma<128,0,0,128,false,false><<<dim3(rt, 8), wblk, 0, stream>>>(
        z1, nullptr, nullptr, nullptr, Wg2, nullptr, h, N);
    selfloop_init_kernel<<<(NF + 255) / 256, 256, 0, stream>>>(h, dinv, z2, NF);
    edge_scatter_kernel<<<eb, 256, 0, stream>>>(src, dst, dinv, h, z2, E);
    bias_relu_kernel<<<(NF + 255) / 256, 256, 0, stream>>>(bg2, z2, NF, 128);

    // z3 = relu(cat(z,z1,z2) @ W2 + b2)   (into h)
    gemm_wmma<128,128,128,128,true,false><<<dim3(rt, 8), wblk, 0, stream>>>(
        z, z1, z2, nullptr, W2, b2, h, N);
    // z4 = relu((z3 + z0) @ W3 + b3)      (into z1 buffer, [N,64])
    gemm_wmma<128,0,0,64,true,true><<<dim3(rt, 4), wblk, 0, stream>>>(
        h, nullptr, nullptr, z0, W3, b3, z1, N);
    // out = z4 @ Wo + bo                  ([N,10])
    gemm_wmma<64,0,0,10,false,false><<<dim3(rt, 1), wblk, 0, stream>>>(
        z1, nullptr, nullptr, nullptr, Wo, bo, (float*)d_out, N);
}